// MSEAndIndexOrderLoss_13348758356601
// MI455X (gfx1250) — compile-verified
//
#include <hip/hip_runtime.h>

typedef float v2f __attribute__((ext_vector_type(2)));
typedef float v8f __attribute__((ext_vector_type(8)));

// Problem constants (from reference setup_inputs / UP_FACTOR)
constexpr int NB   = 128;   // batches
constexpr int WIN  = 128;   // input width
constexpr int WUP  = 1024;  // upsampled width (W * 8)

// Workspace layout (in floats)
constexpr int OFF_T  = 0;                     // tI:   NB*2*WUP = 262144
constexpr int OFF_P  = OFF_T  + NB * 2 * WUP; // pI:   262144
constexpr int OFF_M  = OFF_P  + NB * 2 * WUP; // mI:   NB*WUP = 131072
constexpr int OFF_MD = OFF_M  + NB * WUP;     // minD: 2*NB*WUP = 262144
constexpr int OFF_MI = OFF_MD + 2 * NB * WUP; // minIdx (int): 262144
constexpr int OFF_DP = OFF_MI + 2 * NB * WUP; // direct partials: 512

// ---------------------------------------------------------------------------
// Kernel 1: linear interpolation (align_corners=True) + direct error partials
// ---------------------------------------------------------------------------
__global__ void interp_direct_kernel(const float* __restrict__ tgt,
                                     const float* __restrict__ pred,
                                     const float* __restrict__ mask,
                                     float* __restrict__ ws) {
  const int gid = blockIdx.x * blockDim.x + threadIdx.x;  // exactly NB*WUP threads
  const int n = gid >> 10;
  const int k = gid & (WUP - 1);

  const float pos = (float)k * ((float)(WIN - 1) / (float)(WUP - 1));
  int i0 = (int)pos;
  if (i0 > WIN - 1) i0 = WIN - 1;
  const int i1 = (i0 < WIN - 1) ? i0 + 1 : WIN - 1;
  const float w  = pos - (float)i0;
  const float w0 = 1.0f - w;

  const float* tn = tgt  + n * 2 * WIN;
  const float* pn = pred + n * 2 * WIN;
  const float* mn = mask + n * WIN;

  const float tx = tn[i0] * w0 + tn[i1] * w;
  const float ty = tn[WIN + i0] * w0 + tn[WIN + i1] * w;
  const float px = pn[i0] * w0 + pn[i1] * w;
  const float py = pn[WIN + i0] * w0 + pn[WIN + i1] * w;
  float mv = mn[i0] * w0 + mn[i1] * w;
  mv = (mv < 0.5f) ? 0.0f : mv;

  ws[OFF_T + n * 2 * WUP + k]       = tx;
  ws[OFF_T + n * 2 * WUP + WUP + k] = ty;
  ws[OFF_P + n * 2 * WUP + k]       = px;
  ws[OFF_P + n * 2 * WUP + WUP + k] = py;
  ws[OFF_M + n * WUP + k]           = mv;

  const float dx = tx - px, dy = ty - py;
  float e = (dx * dx + dy * dy) * mv;

  __shared__ float red[256];
  red[threadIdx.x] = e;
  __syncthreads();
  for (int s = 128; s > 0; s >>= 1) {
    if (threadIdx.x < s) red[threadIdx.x] += red[threadIdx.x + s];
    __syncthreads();
  }
  if (threadIdx.x == 0) ws[OFF_DP + blockIdx.x] = red[0];
}

// ---------------------------------------------------------------------------
// Kernel 2: chamfer min/argmin via V_WMMA_F32_16X16X4_F32.
// One block per (dir, n).
//   A row i  = (-2*x2_i, -2*y2_i, |p2_i|^2, 1)
//   B col j  = ( x1_j,    y1_j,   1,        |p1_j|^2)
//   => D[i][j] = squared distance, C = 0.
// B is staged in LDS as two interleaved pair regions so each lane does one
// ds_load_b64 per tile from a uniform per-lane base pointer (no exec masking):
//   region L (lanes 0-15):  (x1_j, y1_j)        = B K0,K1
//   region H (lanes 16-31): (1.0,  |p1_j|^2)    = B K2,K3   (+32 floats pad:
//   the two halves hit disjoint LDS bank halves for conflict-free b64 loads)
// ---------------------------------------------------------------------------
__global__ void chamfer_wmma_kernel(float* __restrict__ ws) {
  constexpr int OFFH = 2 * WUP + 32;
  __shared__ float sP[4 * WUP + 32];

  const int blk = blockIdx.x;
  const int dir = blk >> 7;   // 0: dl(p,t): A=t, B=p ; 1: dl(t,p): A=p, B=t
  const int n   = blk & 127;

  const float* p1 = ws + (dir == 0 ? OFF_P : OFF_T) + n * 2 * WUP;  // points1 (cols)
  const float* p2 = ws + (dir == 0 ? OFF_T : OFF_P) + n * 2 * WUP;  // points2 (rows)

  for (int j = threadIdx.x; j < WUP; j += 256) {
    const float x = p1[j];
    const float y = p1[WUP + j];
    sP[2 * j]            = x;
    sP[2 * j + 1]        = y;
    sP[OFFH + 2 * j]     = 1.0f;
    sP[OFFH + 2 * j + 1] = x * x + y * y;
  }
  __syncthreads();

  const int wave = threadIdx.x >> 5;
  const int lane = threadIdx.x & 31;
  const int lo   = lane & 15;
  const bool hi  = lane >= 16;

  // Uniform per-lane B base: one select, outside all loops.
  const float* bbase = sP + (hi ? OFFH : 0) + 2 * lo;

  float* minD = ws + OFF_MD + (dir * NB + n) * WUP;
  int*   minI = (int*)(ws + OFF_MI) + (dir * NB + n) * WUP;

  for (int it = wave; it < WUP / 16; it += 8) {
    const int irow = it * 16 + lo;
    const float x2 = p2[irow];
    const float y2 = p2[WUP + irow];
    v2f a;
    a.x = hi ? (x2 * x2 + y2 * y2) : (-2.0f * x2);  // lanes 0-15: K0 | 16-31: K2
    a.y = hi ? 1.0f                : (-2.0f * y2);  // lanes 0-15: K1 | 16-31: K3

    float bd[8];
    int   bj[8];
#pragma unroll
    for (int r = 0; r < 8; ++r) { bd[r] = 3.0e38f; bj[r] = 0; }

    // Software pipeline: prefetch next tile's B while the current WMMA runs.
    v2f bcur = *(const v2f*)(bbase);  // tile jt = 0
    for (int jt = 0; jt < WUP / 16; ++jt) {
      const int jn = (jt + 1) & (WUP / 16 - 1);   // branchless wrap
      const v2f bnext = *(const v2f*)(bbase + 32 * jn);

      v8f c = {};
      c = __builtin_amdgcn_wmma_f32_16x16x4_f32(
          /*neg_a=*/false, a, /*neg_b=*/false, bcur,
          /*c_mod=*/(short)0, c, /*reuse_a=*/false, /*reuse_b=*/false);

      const int j = jt * 16 + lo;
#pragma unroll
      for (int r = 0; r < 8; ++r) {
        const float d = c[r];
        const bool u = d < bd[r];      // strict < keeps first occurrence (j grows)
        bd[r] = u ? d : bd[r];
        bj[r] = u ? j : bj[r];
      }
      bcur = bnext;
    }

    // Reduce min over the 16 columns held across each 16-lane half.
#pragma unroll
    for (int off = 8; off >= 1; off >>= 1) {
#pragma unroll
      for (int r = 0; r < 8; ++r) {
        const float od = __shfl_xor(bd[r], off, 32);
        const int   oj = __shfl_xor(bj[r], off, 32);
        const bool u = (od < bd[r]) || (od == bd[r] && oj < bj[r]);
        bd[r] = u ? od : bd[r];
        bj[r] = u ? oj : bj[r];
      }
    }

    if (lo == 0) {
      const int base = it * 16 + (hi ? 8 : 0);  // lanes 16-31 hold rows M=8..15
#pragma unroll
      for (int r = 0; r < 8; ++r) {
        minD[base + r] = bd[r];
        minI[base + r] = bj[r];
      }
    }
  }
}

// ---------------------------------------------------------------------------
// Kernel 3: deterministic single-block finalize.
// ---------------------------------------------------------------------------
__global__ void finalize_kernel(const float* __restrict__ ws,
                                float* __restrict__ out) {
  __shared__ float r0[256];
  __shared__ float r1[256];

  float sdir = 0.0f;
  for (int b = threadIdx.x; b < 512; b += 256) sdir += ws[OFF_DP + b];

  float smse = 0.0f, spen = 0.0f;
  const float* minD = ws + OFF_MD;
  const int*   minI = (const int*)(ws + OFF_MI);
  const float* mI   = ws + OFF_M;
  for (int f = threadIdx.x; f < 2 * NB * WUP; f += 256) {
    const int i  = f & (WUP - 1);
    const int dn = f >> 10;
    const int n  = dn & (NB - 1);
    const float m = mI[n * WUP + i];
    smse += minD[f] * m;
    if (i < WUP - 1) {
      const float diff = -(float)(minI[f + 1] - minI[f]);
      if (diff > 0.0f) spen += diff * diff * m;
    }
  }

  r0[threadIdx.x] = sdir;
  __syncthreads();
  for (int s = 128; s > 0; s >>= 1) {
    if (threadIdx.x < s) r0[threadIdx.x] += r0[threadIdx.x + s];
    __syncthreads();
  }
  const float directSum = r0[0];
  __syncthreads();

  r0[threadIdx.x] = smse;
  r1[threadIdx.x] = spen;
  __syncthreads();
  for (int s = 128; s > 0; s >>= 1) {
    if (threadIdx.x < s) {
      r0[threadIdx.x] += r0[threadIdx.x + s];
      r1[threadIdx.x] += r1[threadIdx.x + s];
    }
    __syncthreads();
  }

  if (threadIdx.x == 0) {
    const float direct_mean = directSum / (float)(NB * WUP);
    // matched = 0.5*(mse0+mse1 means) + 0.5*0.1*(pen0+pen1 means)
    const float matched = r0[0] / (2.0f * (float)(NB * WUP)) +
                          0.1f * r1[0] / (2.0f * (float)(NB * (WUP - 1)));
    out[0] = fminf(matched, direct_mean);
  }
}

// ---------------------------------------------------------------------------
extern "C" void kernel_launch(void* const* d_in, const int* in_sizes, int n_in,
                              void* d_out, int out_size, void* d_ws, size_t ws_size,
                              hipStream_t stream) {
  (void)in_sizes; (void)n_in; (void)out_size; (void)ws_size;
  const float* tgt  = (const float*)d_in[0];  // target_points (128,2,128)
  const float* pred = (const float*)d_in[1];  // predictions   (128,2,128)
  const float* mask = (const float*)d_in[2];  // visibility_mask (128,128)
  float* ws  = (float*)d_ws;
  float* out = (float*)d_out;

  interp_direct_kernel<<<(NB * WUP) / 256, 256, 0, stream>>>(tgt, pred, mask, ws);
  chamfer_wmma_kernel<<<2 * NB, 256, 0, stream>>>(ws);
  finalize_kernel<<<1, 256, 0, stream>>>(ws, out);
}